// SAGE_90726889160780
// MI455X (gfx1250) — compile-verified
//
#include <hip/hip_runtime.h>
#include <hip/hip_bf16.h>
#include <math.h>

// ---------------------------------------------------------------------------
// GraphSAGE, 3 layers, on gfx1250 (MI455X).
// Phase 1 (per layer): edge scatter-add (atomic f32, L2-resident), mean via
//   precomputed 1/deg (dst is layer-invariant). Phase 2: fused
//   (agg*inv)@Wl + b + h@Wr with fp32-native V_WMMA_F32_16X16X4_F32.
//   FOUT/ELU are template params so all weight loads use immediate offsets.
// ---------------------------------------------------------------------------

typedef __attribute__((ext_vector_type(2))) float v2f;
typedef __attribute__((ext_vector_type(8))) float v8f;

#define FIN 128

// ----------------------------- utility kernels -----------------------------

__global__ void zero_f32(float* __restrict__ p, int n) {
    int i = blockIdx.x * blockDim.x + threadIdx.x;
    if (i < n) p[i] = 0.0f;
}

__global__ void degree_kernel(const int* __restrict__ dst, float* __restrict__ cnt,
                              int nedges) {
    int e = blockIdx.x * blockDim.x + threadIdx.x;
    if (e < nedges) atomicAdd(&cnt[dst[e]], 1.0f);
}

__global__ void invdeg_kernel(const float* __restrict__ cnt, float* __restrict__ inv,
                              int n) {
    int i = blockIdx.x * blockDim.x + threadIdx.x;
    if (i < n) inv[i] = 1.0f / fmaxf(cnt[i], 1.0f);
}

// One wave per edge: lane loads float4 (feat lane*4..lane*4+3) -> 4 atomics.
__global__ void scatter_add_kernel(const float* __restrict__ h,
                                   const int* __restrict__ src,
                                   const int* __restrict__ dst,
                                   float* __restrict__ agg, int nedges) {
    int e    = (blockIdx.x * blockDim.x + threadIdx.x) >> 5;
    int lane = threadIdx.x & 31;
    if (e >= nedges) return;
    int s = src[e];
    int d = dst[e];
    const float4 v = ((const float4*)(h + (long)s * FIN))[lane];
    float* ap = agg + (long)d * FIN + lane * 4;
    atomicAdd(ap + 0, v.x);
    atomicAdd(ap + 1, v.y);
    atomicAdd(ap + 2, v.z);
    atomicAdd(ap + 3, v.w);
}

// ------------------------------- WMMA GEMM --------------------------------
// Block: 16 output rows, blockDim = 32 * (FOUT/16) waves; wave w owns columns
// [w*16, w*16+16). K = 128 (agg @ Wl) + 128 (h @ Wr), fp32 WMMA 16x16x4.
// A layout (16x4 f32): lane holds row lane%16; vgpr0/1 = K { k+2*(lane/16),
// k+2*(lane/16)+1 }. B symmetric with columns across lanes.
// C/D: c[i] = D[row = i + 8*(lane/16)][col = lane%16].

template <int FOUT, bool ELU>
__global__ void __launch_bounds__(32 * (FOUT / 16))
sage_gemm_kernel(const float* __restrict__ agg, const float* __restrict__ invdeg,
                 const float* __restrict__ h,
                 const float* __restrict__ Wl, const float* __restrict__ bl,
                 const float* __restrict__ Wr, float* __restrict__ out) {
    __shared__ float As[16][132];   // agg * inv_deg  (pad 132: conflict-free)
    __shared__ float Xs[16][132];   // h

    const int m0 = blockIdx.x * 16;

    // Stage 16x128 tiles into LDS as float4 (row stride 528B is 16B-aligned).
    for (int i = threadIdx.x; i < 16 * (FIN / 4); i += blockDim.x) {
        int r  = i >> 5;           // /32 float4s per row
        int c4 = i & 31;
        long g = (long)(m0 + r) * (FIN / 4) + c4;
        float inv = invdeg[m0 + r];
        float4 va = ((const float4*)agg)[g];
        float4 vx = ((const float4*)h)[g];
        va.x *= inv; va.y *= inv; va.z *= inv; va.w *= inv;
        *(float4*)&As[r][c4 * 4] = va;
        *(float4*)&Xs[r][c4 * 4] = vx;
    }
    __syncthreads();

    const int lane = threadIdx.x & 31;
    const int wave = threadIdx.x >> 5;
    const int n0   = wave * 16;
    const int row  = lane & 15;   // A row within tile
    const int col  = lane & 15;   // B/C column within tile
    const int hi   = lane >> 4;   // K-pair select (0: K0/K1, 1: K2/K3)

    // Per-wave weight base pointers; all k-offsets below are compile-time
    // immediates (max 125*FOUT*4 bytes < 2^23), so no per-iteration address math.
    const float* wl = Wl + 2 * hi * FOUT + n0 + col;
    const float* wr = Wr + 2 * hi * FOUT + n0 + col;

    v8f c = {};

    // Phase 1: (agg/deg) @ Wl
#pragma unroll
    for (int k = 0; k < FIN; k += 4) {
        v2f a, b;
        a.x = As[row][k + 2 * hi];
        a.y = As[row][k + 2 * hi + 1];
        b.x = wl[k * FOUT];
        b.y = wl[(k + 1) * FOUT];
        c = __builtin_amdgcn_wmma_f32_16x16x4_f32(false, a, false, b,
                                                  (short)0, c, false, false);
    }
    // Phase 2: h @ Wr (accumulate into same C)
#pragma unroll
    for (int k = 0; k < FIN; k += 4) {
        v2f a, b;
        a.x = Xs[row][k + 2 * hi];
        a.y = Xs[row][k + 2 * hi + 1];
        b.x = wr[k * FOUT];
        b.y = wr[(k + 1) * FOUT];
        c = __builtin_amdgcn_wmma_f32_16x16x4_f32(false, a, false, b,
                                                  (short)0, c, false, false);
    }

    const float bias = bl[n0 + col];
#pragma unroll
    for (int i = 0; i < 8; ++i) {
        float v = c[i] + bias;
        if (ELU) v = (v > 0.0f) ? v : expm1f(v);   // ELU(alpha=1)
        out[(long)(m0 + i + 8 * hi) * FOUT + n0 + col] = v;
    }
}

// ------------------------------ log_softmax -------------------------------
// One wave per row of 64; lane owns elements {lane, lane+32}. In-place.
__global__ void logsoftmax64_kernel(float* __restrict__ out, int n) {
    int row  = (blockIdx.x * blockDim.x + threadIdx.x) >> 5;
    int lane = threadIdx.x & 31;
    if (row >= n) return;
    float* p = out + (long)row * 64;
    float a = p[lane];
    float b = p[lane + 32];
    float m = fmaxf(a, b);
#pragma unroll
    for (int o = 16; o > 0; o >>= 1) m = fmaxf(m, __shfl_xor(m, o, 32));
    float s = __expf(a - m) + __expf(b - m);
#pragma unroll
    for (int o = 16; o > 0; o >>= 1) s += __shfl_xor(s, o, 32);
    float l = m + __logf(s);
    p[lane]      = a - l;
    p[lane + 32] = b - l;
}

// -------------------------------- launcher --------------------------------

extern "C" void kernel_launch(void* const* d_in, const int* in_sizes, int n_in,
                              void* d_out, int out_size, void* d_ws, size_t ws_size,
                              hipStream_t stream) {
    const float* x   = (const float*)d_in[0];
    const int*   ei  = (const int*)d_in[1];     // edge_index [2, E]
    const float* Wl0 = (const float*)d_in[2];
    const float* bl0 = (const float*)d_in[3];
    const float* Wr0 = (const float*)d_in[4];
    const float* Wl1 = (const float*)d_in[5];
    const float* bl1 = (const float*)d_in[6];
    const float* Wr1 = (const float*)d_in[7];
    const float* Wl2 = (const float*)d_in[8];
    const float* bl2 = (const float*)d_in[9];
    const float* Wr2 = (const float*)d_in[10];

    const int n_nodes = in_sizes[0] / FIN;      // 50000 (divisible by 16)
    const int n_edges = in_sizes[1] / 2;        // 800000
    const int* src = ei;
    const int* dst = ei + n_edges;

    // Workspace layout (floats): agg[N*128] | hbuf[N*128] | cnt[N] | inv[N]
    float* agg  = (float*)d_ws;
    float* hbuf = agg + (long)n_nodes * FIN;
    float* cnt  = hbuf + (long)n_nodes * FIN;
    float* inv  = cnt + n_nodes;

    const int nf     = n_nodes * FIN;
    const int zb     = (nf + 255) / 256;
    const int eb1    = (n_edges + 255) / 256;          // 1 thread / edge
    const int ebw    = (n_edges * 32 + 255) / 256;     // 1 wave / edge
    const int nb     = (n_nodes + 255) / 256;
    const int mtiles = n_nodes / 16;

    // Degree (dst identical across layers) -> inv_deg, once.
    zero_f32<<<nb, 256, 0, stream>>>(cnt, n_nodes);
    degree_kernel<<<eb1, 256, 0, stream>>>(dst, cnt, n_edges);
    invdeg_kernel<<<nb, 256, 0, stream>>>(cnt, inv, n_nodes);

    // ---- Layer 0: x -> hbuf (128->128, ELU) ----
    zero_f32<<<zb, 256, 0, stream>>>(agg, nf);
    scatter_add_kernel<<<ebw, 256, 0, stream>>>(x, src, dst, agg, n_edges);
    sage_gemm_kernel<128, true><<<mtiles, 256, 0, stream>>>(agg, inv, x,
                                                            Wl0, bl0, Wr0, hbuf);

    // ---- Layer 1: hbuf -> hbuf (in-place safe: block stages its own 16 rows
    //      into LDS before writing them) ----
    zero_f32<<<zb, 256, 0, stream>>>(agg, nf);
    scatter_add_kernel<<<ebw, 256, 0, stream>>>(hbuf, src, dst, agg, n_edges);
    sage_gemm_kernel<128, true><<<mtiles, 256, 0, stream>>>(agg, inv, hbuf,
                                                            Wl1, bl1, Wr1, hbuf);

    // ---- Layer 2: hbuf -> d_out (128->64, no ELU), then log_softmax ----
    zero_f32<<<zb, 256, 0, stream>>>(agg, nf);
    scatter_add_kernel<<<ebw, 256, 0, stream>>>(hbuf, src, dst, agg, n_edges);
    sage_gemm_kernel<64, false><<<mtiles, 128, 0, stream>>>(agg, inv, hbuf,
                                                            Wl2, bl2, Wr2,
                                                            (float*)d_out);

    logsoftmax64_kernel<<<(n_nodes * 32 + 255) / 256, 256, 0, stream>>>(
        (float*)d_out, n_nodes);
}